// MNN_35708358099297
// MI455X (gfx1250) — compile-verified
//
#include <hip/hip_runtime.h>

typedef __attribute__((ext_vector_type(2))) float v2f;
typedef __attribute__((ext_vector_type(4))) float v4f;
typedef __attribute__((ext_vector_type(8))) float v8f;

constexpr int   N1   = 8192;   // rows per half
constexpr int   DIM  = 64;
constexpr float FBIG = 3.402823466e+38f;

// Insert v into ascending sorted 5-list (t0 <= t1 <= ... <= t4). t4 = current 5th smallest.
__device__ __forceinline__ void top5_insert(float v, float& t0, float& t1, float& t2,
                                            float& t3, float& t4) {
  if (v < t4) {
    if (v < t3) {
      t4 = t3;
      if (v < t2) {
        t3 = t2;
        if (v < t1) {
          t2 = t1;
          if (v < t0) { t1 = t0; t0 = v; } else { t1 = v; }
        } else { t2 = v; }
      } else { t3 = v; }
    } else { t4 = v; }
  }
}

// Compute two 16x16 tiles of A·B^T (columns jt*16.. and (jt+1)*16..) into c0/c1.
// All 32 B-operand float2s are preloaded so the backend can clause the loads and
// interleave two independent WMMA accumulation chains.
__device__ __forceinline__ void dot_two_tiles(const float* __restrict__ B, int jt,
                                              int lm, int half, const v2f* areg,
                                              v8f& c0, v8f& c1) {
  const float* p0 = B + (size_t)(jt * 16 + lm) * DIM + 2 * half;
  const float* p1 = p0 + 16 * DIM;
  v2f b0[16], b1[16];
#pragma unroll
  for (int s = 0; s < 16; ++s) b0[s] = *(const v2f*)(p0 + 4 * s);
#pragma unroll
  for (int s = 0; s < 16; ++s) b1[s] = *(const v2f*)(p1 + 4 * s);
  c0 = {};
  c1 = {};
#pragma unroll
  for (int s = 0; s < 16; ++s) {
    c0 = __builtin_amdgcn_wmma_f32_16x16x4_f32(false, areg[s], false, b0[s],
                                               (short)0, c0, false, false);
    c1 = __builtin_amdgcn_wmma_f32_16x16x4_f32(false, areg[s], false, b1[s],
                                               (short)0, c1, false, false);
  }
}

// ---------------------------------------------------------------------------
// Row squared-norms of x (2*N1 rows, DIM=64 cols).
// ---------------------------------------------------------------------------
__global__ __launch_bounds__(256) void rownorm_kernel(const float* __restrict__ x,
                                                      float* __restrict__ sq) {
  const int row = blockIdx.x * 256 + threadIdx.x;
  const v4f* p = (const v4f*)(x + (size_t)row * DIM);
  float s = 0.0f;
#pragma unroll
  for (int k = 0; k < DIM / 4; ++k) {
    v4f v = p[k];
    s += v.x * v.x + v.y * v.y + v.z * v.z + v.w * v.w;
  }
  sq[row] = s;
}

__global__ void zero_acc_kernel(float* __restrict__ acc) {
  if (threadIdx.x == 0) acc[0] = 0.0f;
}

// ---------------------------------------------------------------------------
// Pass 1: for each row i of A (one 16-row tile per 32-thread block), compute
// kth_out[i] = 5th smallest of d(i, :) over all N1 columns of B.
// d(i,j) = sqA[i] + sqB[j] - 2 * dot(A_i, B_j), via chained f32 WMMA.
// ---------------------------------------------------------------------------
__global__ __launch_bounds__(32) void knn_kth_kernel(const float* __restrict__ A,
                                                     const float* __restrict__ B,
                                                     const float* __restrict__ sqA,
                                                     const float* __restrict__ sqB,
                                                     float* __restrict__ kth_out) {
  __shared__ float lds[16][33];   // 16 rows x 32 cols (two tiles), padded
  const int lane = threadIdx.x & 31;
  const int half = lane >> 4;     // K-pair group for A/B operands; M-group for C
  const int lm   = lane & 15;
  const int rowBase = blockIdx.x * 16;

  // Preload A tile: lane holds row (rowBase+lm), float2 at k = 4*s + 2*half.
  v2f areg[16];
#pragma unroll
  for (int s = 0; s < 16; ++s)
    areg[s] = *(const v2f*)(A + (size_t)(rowBase + lm) * DIM + 4 * s + 2 * half);

  float sqa[8];
#pragma unroll
  for (int r = 0; r < 8; ++r) sqa[r] = sqA[rowBase + r + 8 * half];

  float t0 = FBIG, t1 = FBIG, t2 = FBIG, t3 = FBIG, t4 = FBIG;

  for (int jt = 0; jt < N1 / 16; jt += 2) {
    v8f c0, c1;
    dot_two_tiles(B, jt, lm, half, areg, c0, c1);
    const float sqb0 = sqB[jt * 16 + lm];
    const float sqb1 = sqB[(jt + 1) * 16 + lm];
#pragma unroll
    for (int r = 0; r < 8; ++r) {
      lds[r + 8 * half][lm]      = sqa[r] + sqb0 - 2.0f * c0[r];
      lds[r + 8 * half][16 + lm] = sqa[r] + sqb1 - 2.0f * c1[r];
    }
    __syncthreads();
    // Each lane scans 16 values of row lm (half selects column sub-range).
#pragma unroll
    for (int n = 0; n < 16; ++n)
      top5_insert(lds[lm][half * 16 + n], t0, t1, t2, t3, t4);
    __syncthreads();
  }

  // Merge the two half-wave lists for each row.
  const float u0 = __shfl(t0, lane + 16, 32);
  const float u1 = __shfl(t1, lane + 16, 32);
  const float u2 = __shfl(t2, lane + 16, 32);
  const float u3 = __shfl(t3, lane + 16, 32);
  const float u4 = __shfl(t4, lane + 16, 32);
  if (lane < 16) {
    top5_insert(u0, t0, t1, t2, t3, t4);
    top5_insert(u1, t0, t1, t2, t3, t4);
    top5_insert(u2, t0, t1, t2, t3, t4);
    top5_insert(u3, t0, t1, t2, t3, t4);
    top5_insert(u4, t0, t1, t2, t3, t4);
    kth_out[rowBase + lane] = t4;
  }
}

// ---------------------------------------------------------------------------
// Pass 2: sum d(i,j) over entries with d <= kth_row[i] && d <= kth_col[j].
// Column range split into 8 chunks of 64 tiles for occupancy.
// ---------------------------------------------------------------------------
__global__ __launch_bounds__(32) void mnn_sum_kernel(const float* __restrict__ A,
                                                     const float* __restrict__ B,
                                                     const float* __restrict__ sqA,
                                                     const float* __restrict__ sqB,
                                                     const float* __restrict__ kthA,
                                                     const float* __restrict__ kthB,
                                                     float* __restrict__ acc) {
  const int lane  = threadIdx.x & 31;
  const int half  = lane >> 4;
  const int lm    = lane & 15;
  const int rt    = blockIdx.x & (N1 / 16 - 1);  // 512 row tiles
  const int chunk = blockIdx.x >> 9;             // 8 column chunks
  const int rowBase = rt * 16;

  v2f areg[16];
#pragma unroll
  for (int s = 0; s < 16; ++s)
    areg[s] = *(const v2f*)(A + (size_t)(rowBase + lm) * DIM + 4 * s + 2 * half);

  float sqa[8], ka[8];
#pragma unroll
  for (int r = 0; r < 8; ++r) {
    sqa[r] = sqA[rowBase + r + 8 * half];
    ka[r]  = kthA[rowBase + r + 8 * half];
  }

  float psum = 0.0f;
  const int jt0 = chunk * 64, jt1 = jt0 + 64;
  for (int jt = jt0; jt < jt1; jt += 2) {
    v8f c0, c1;
    dot_two_tiles(B, jt, lm, half, areg, c0, c1);
    const float sqb0 = sqB[jt * 16 + lm];
    const float sqb1 = sqB[(jt + 1) * 16 + lm];
    const float kb0  = kthB[jt * 16 + lm];
    const float kb1  = kthB[(jt + 1) * 16 + lm];
#pragma unroll
    for (int r = 0; r < 8; ++r) {
      const float d0 = sqa[r] + sqb0 - 2.0f * c0[r];
      const float d1 = sqa[r] + sqb1 - 2.0f * c1[r];
      if (d0 <= ka[r] && d0 <= kb0) psum += d0;
      if (d1 <= ka[r] && d1 <= kb1) psum += d1;
    }
  }

  // Wave32 reduction.
#pragma unroll
  for (int o = 16; o > 0; o >>= 1) psum += __shfl_down(psum, o, 32);
  if (lane == 0) atomicAdd(acc, psum);
}

__global__ void finalize_kernel(const float* __restrict__ acc, float* __restrict__ out) {
  if (threadIdx.x == 0)
    out[0] = acc[0] * (1.0f / ((float)N1 * (float)N1));  // LAMBDA_REG = 1
}

// ---------------------------------------------------------------------------
extern "C" void kernel_launch(void* const* d_in, const int* in_sizes, int n_in,
                              void* d_out, int out_size, void* d_ws, size_t ws_size,
                              hipStream_t stream) {
  const float* x  = (const float*)d_in[0];   // (16384, 64) f32; d_in[1] ('b') unused
  const float* x0 = x;
  const float* x1 = x + (size_t)N1 * DIM;

  float* sq      = (float*)d_ws;        // 2*N1 row norms
  float* sq0     = sq;
  float* sq1     = sq + N1;
  float* kth_row = sq + 2 * N1;         // N1
  float* kth_col = kth_row + N1;        // N1
  float* acc     = kth_col + N1;        // 1

  rownorm_kernel<<<(2 * N1) / 256, 256, 0, stream>>>(x, sq);
  zero_acc_kernel<<<1, 32, 0, stream>>>(acc);
  // Row-wise 5th smallest of d.
  knn_kth_kernel<<<N1 / 16, 32, 0, stream>>>(x0, x1, sq0, sq1, kth_row);
  // Column-wise 5th smallest of d == row-wise of d^T (operands swapped).
  knn_kth_kernel<<<N1 / 16, 32, 0, stream>>>(x1, x0, sq1, sq0, kth_col);
  // Masked sum.
  mnn_sum_kernel<<<(N1 / 16) * 8, 32, 0, stream>>>(x0, x1, sq0, sq1,
                                                   kth_row, kth_col, acc);
  finalize_kernel<<<1, 32, 0, stream>>>(acc, (float*)d_out);
}